// DAGExecutor_76630806495353
// MI455X (gfx1250) — compile-verified
//
#include <hip/hip_runtime.h>
#include <stdint.h>

// Problem constants (reference: B=8, T=4096, DAG_DEPTH=16, TOTAL_NODES=32)
#define DEPTH 16
#define NODES 32
#define RING  6            // async prefetch depth (steps); 6*8=48 outstanding <= 63 (ASYNCcnt)
#define BLOCK 32           // one wave per block -> max spread over WGPs
#define LOG_LIM 100.0f
#define INV_TEMP 10000.0f  // 1/TEMP, TEMP = 1e-4

typedef float v2f __attribute__((ext_vector_type(2)));

// tanh(x * 1e4) via exact algebraic identity, hw v_exp_f32 + v_rcp_f32.
__device__ __forceinline__ float fast_tanh_scaled(float x) {
    const float y = x * INV_TEMP;
    const float t = __expf(-2.0f * fabsf(y));                       // ->0 for |y|>~44
    const float r = (1.0f - t) * __builtin_amdgcn_rcpf(1.0f + t);   // (1-t)/(1+t)
    return copysignf(r, y);
}

// Stage O[p, T, 0..31] (128B per thread, 4KB per wave) into ring slot T%RING.
// Chunk-major transposed layout: issue i -> LDS slot + i*512 + lane*16
//  * async write per issue: 512B contiguous (32 lanes x b128)
//  * compute read per chunk: lane-contiguous 512B ds_load_b128, conflict-free
template<int T>
__device__ __forceinline__ void issue_group(uint64_t gO, unsigned ldsLane) {
    if constexpr (T < DEPTH) {
        const unsigned slotBase = (unsigned)((T % RING) * 4096);
        const uint64_t gstep    = gO + (uint64_t)(T * 128);
#pragma unroll
        for (int i = 0; i < 8; ++i) {
            unsigned laddr = ldsLane + slotBase + (unsigned)(i * 512);
            uint64_t gaddr = gstep + (uint64_t)(i * 16);
            asm volatile("global_load_async_to_lds_b128 %0, %1, off"
                         :: "v"(laddr), "v"(gaddr)
                         : "memory");
        }
    }
}

template<int N>
__device__ __forceinline__ void wait_async() {
    asm volatile("s_wait_asynccnt %0" :: "i"(N) : "memory");
}

template<int S>
__device__ __forceinline__ void dag_step(const float* __restrict__ ldsB, int lane,
                                         v2f (&lm2)[NODES / 2],   // packed (logm[2k], logm[2k+1])
                                         v2f (&sg2)[NODES / 2],   // packed (sgn[2k],  sgn[2k+1])
                                         float (&vs2)[NODES],     // 2*Vs[n]
                                         const float (&gA)[DEPTH],
                                         uint64_t gO, unsigned ldsLane, float& fin)
{
    constexpr int NACT = 16 + S;                 // causal: nodes >= 16+S never contribute
    constexpr int NCH  = (NACT + 3) / 4;
    // groups issued so far: 0 .. min(15, S+RING-1); require group S complete.
    constexpr int WAITN = 8 * ((((S + RING) < DEPTH) ? (S + RING) : DEPTH) - S - 1);
    wait_async<WAITN>();

    const float4* slotp = reinterpret_cast<const float4*>(ldsB + (S % RING) * 1024);

    v2f A0 = {0.f, 0.f}, A1 = {0.f, 0.f};        // packed dot accumulators (even,odd)
    v2f B0 = {0.f, 0.f}, B1 = {0.f, 0.f};
    float P0 = 1.f, P1 = 1.f, P2 = 1.f, P3 = 1.f;
#pragma unroll
    for (int c = 0; c < NCH; ++c) {
        const float4 ov = slotp[c * 32 + lane];                 // ds_load_b128
        const v2f olo = {ov.x, ov.y};                           // subregister pairs
        const v2f ohi = {ov.z, ov.w};
        const float oe[4] = {ov.x, ov.y, ov.z, ov.w};
        const int k0 = c * 2, k1 = c * 2 + 1;

        // ---- packed dots: one v_pk_fma_f32 per (pair, dot) ----
        if (c * 4 + 1 < NACT) {                                 // low pair full
            if (c & 1) {
                A1 = __builtin_elementwise_fma(olo, lm2[k0], A1);
                B1 = __builtin_elementwise_fma(olo, sg2[k0], B1);
            } else {
                A0 = __builtin_elementwise_fma(olo, lm2[k0], A0);
                B0 = __builtin_elementwise_fma(olo, sg2[k0], B0);
            }
        } else if (c * 4 < NACT) {                              // lone node 4c (NACT%4==1)
            A0.x = fmaf(ov.x, lm2[k0].x, A0.x);
            B0.x = fmaf(ov.x, sg2[k0].x, B0.x);
        }
        if (c * 4 + 3 < NACT) {                                 // high pair full
            if (c & 1) {
                A0 = __builtin_elementwise_fma(ohi, lm2[k1], A0);
                B0 = __builtin_elementwise_fma(ohi, sg2[k1], B0);
            } else {
                A1 = __builtin_elementwise_fma(ohi, lm2[k1], A1);
                B1 = __builtin_elementwise_fma(ohi, sg2[k1], B1);
            }
        } else if (c * 4 + 2 < NACT) {                          // lone node 4c+2 (NACT%4==3)
            A1.x = fmaf(ov.z, lm2[k1].x, A1.x);
            B1.x = fmaf(ov.z, sg2[k1].x, B1.x);
        }

        // ---- sign-weight product: fma-with-|abs|-modifier + mul per node ----
#pragma unroll
        for (int j = 0; j < 4; ++j) {
            const int n = c * 4 + j;
            if (n < NACT) {                                     // static after unroll
                if (j == 0)      P0 *= fmaf(vs2[n], fabsf(oe[j]), 1.0f);
                else if (j == 1) P1 *= fmaf(vs2[n], fabsf(oe[j]), 1.0f);
                else if (j == 2) P2 *= fmaf(vs2[n], fabsf(oe[j]), 1.0f);
                else             P3 *= fmaf(vs2[n], fabsf(oe[j]), 1.0f);
            }
        }
    }

    const float A  = (A0.x + A0.y) + (A1.x + A1.y);
    const float Bv = (B0.x + B0.y) + (B1.x + B1.y);
    const float P  = (P0 * P1) * (P2 * P3);

    const float g  = gA[S];
    const float gi = 1.0f - g;
    const float R  = gi * A + g * Bv;            // (1-g)*dot(O,logm)+g*dot(O,signed)

    const float linear_sign = fast_tanh_scaled(R);
    const float log_sign    = fast_tanh_scaled(P);
    const float Vs_new = g * linear_sign + gi * log_sign;
    const float Vm_new = g * fabsf(R) + gi * __expf(fminf(R, LOG_LIM));

    if constexpr (S < DEPTH - 1) {
        constexpr int K = (16 + S) / 2;
        const float lmNew = __logf(fmaxf(Vm_new, 1e-12f));      // one v_log per step total
        const float sgNew = Vs_new * Vm_new;
        if constexpr ((S & 1) == 0) { lm2[K].x = lmNew; sg2[K].x = sgNew; }
        else                        { lm2[K].y = lmNew; sg2[K].y = sgNew; }
        vs2[16 + S] = 2.0f * Vs_new;
        issue_group<S + RING>(gO, ldsLane);                     // refill the slot just consumed
    } else {
        fin = Vs_new * Vm_new;                                  // final = Vs[31]*Vm[31]
    }
}

__global__ __launch_bounds__(BLOCK) void dag_exec_kernel(
    const float* __restrict__ Vmag,   // [BT, 32]
    const float* __restrict__ Vsign,  // [BT, 32]
    const float* __restrict__ O,      // [BT, 16, 32]
    const float* __restrict__ G,      // [BT, 16]
    float* __restrict__ out,          // [BT]
    int total)
{
    __shared__ float lds[RING * 1024];            // 24KB ring: RING slots x 4KB
    const int lane = threadIdx.x;                 // 0..31, one wave per block
    const int p    = blockIdx.x * BLOCK + lane;   // this thread's (b,t)
    if (p >= total) return;                       // total % 32 == 0 -> wave-uniform

    const uint64_t gO      = (uint64_t)(uintptr_t)O + (uint64_t)p * (DEPTH * NODES * 4);
    const unsigned ldsLane = (unsigned)(uintptr_t)(void*)&lds[0] + (unsigned)(lane * 16);

    // Prologue: prefetch steps 0..RING-1 (48 async b128 issues in flight)
    issue_group<0>(gO, ldsLane);
    issue_group<1>(gO, ldsLane);
    issue_group<2>(gO, ldsLane);
    issue_group<3>(gO, ldsLane);
    issue_group<4>(gO, ldsLane);
    issue_group<5>(gO, ldsLane);

    // Node state (registers, fully static indexing). Nodes >= 16 are write-before-read:
    // the causal mask guarantees their initial values are never consumed.
    v2f   lm2[NODES / 2], sg2[NODES / 2];
    float vs2[NODES], gA[DEPTH];
    {
        const float4* vm4 = reinterpret_cast<const float4*>(Vmag  + (size_t)p * NODES);
        const float4* vs4 = reinterpret_cast<const float4*>(Vsign + (size_t)p * NODES);
        const float4* g4  = reinterpret_cast<const float4*>(G     + (size_t)p * DEPTH);
#pragma unroll
        for (int c = 0; c < 4; ++c) {
            const float4 m  = vm4[c];
            const float4 sv = vs4[c];
            const float4 gg = g4[c];
            const float me[4] = {m.x, m.y, m.z, m.w};
            const float se[4] = {sv.x, sv.y, sv.z, sv.w};
            const float ge[4] = {gg.x, gg.y, gg.z, gg.w};
            float lme[4], sge[4];
#pragma unroll
            for (int j = 0; j < 4; ++j) {
                const int n = c * 4 + j;
                lme[j] = __logf(fmaxf(me[j], 1e-12f));
                sge[j] = se[j] * me[j];
                vs2[n] = 2.0f * se[j];
                gA[n]  = ge[j];
            }
            lm2[c * 2]     = (v2f){lme[0], lme[1]};
            lm2[c * 2 + 1] = (v2f){lme[2], lme[3]};
            sg2[c * 2]     = (v2f){sge[0], sge[1]};
            sg2[c * 2 + 1] = (v2f){sge[2], sge[3]};
        }
    }

    float fin = 0.0f;
    const float* ldsB = &lds[0];
    dag_step< 0>(ldsB, lane, lm2, sg2, vs2, gA, gO, ldsLane, fin);
    dag_step< 1>(ldsB, lane, lm2, sg2, vs2, gA, gO, ldsLane, fin);
    dag_step< 2>(ldsB, lane, lm2, sg2, vs2, gA, gO, ldsLane, fin);
    dag_step< 3>(ldsB, lane, lm2, sg2, vs2, gA, gO, ldsLane, fin);
    dag_step< 4>(ldsB, lane, lm2, sg2, vs2, gA, gO, ldsLane, fin);
    dag_step< 5>(ldsB, lane, lm2, sg2, vs2, gA, gO, ldsLane, fin);
    dag_step< 6>(ldsB, lane, lm2, sg2, vs2, gA, gO, ldsLane, fin);
    dag_step< 7>(ldsB, lane, lm2, sg2, vs2, gA, gO, ldsLane, fin);
    dag_step< 8>(ldsB, lane, lm2, sg2, vs2, gA, gO, ldsLane, fin);
    dag_step< 9>(ldsB, lane, lm2, sg2, vs2, gA, gO, ldsLane, fin);
    dag_step<10>(ldsB, lane, lm2, sg2, vs2, gA, gO, ldsLane, fin);
    dag_step<11>(ldsB, lane, lm2, sg2, vs2, gA, gO, ldsLane, fin);
    dag_step<12>(ldsB, lane, lm2, sg2, vs2, gA, gO, ldsLane, fin);
    dag_step<13>(ldsB, lane, lm2, sg2, vs2, gA, gO, ldsLane, fin);
    dag_step<14>(ldsB, lane, lm2, sg2, vs2, gA, gO, ldsLane, fin);
    dag_step<15>(ldsB, lane, lm2, sg2, vs2, gA, gO, ldsLane, fin);

    out[p] = fin;                                 // coalesced b32 store
}

extern "C" void kernel_launch(void* const* d_in, const int* in_sizes, int n_in,
                              void* d_out, int out_size, void* d_ws, size_t ws_size,
                              hipStream_t stream) {
    const float* Vmag  = (const float*)d_in[0];  // (8,4096,32) f32
    const float* Vsign = (const float*)d_in[1];  // (8,4096,32) f32
    const float* O     = (const float*)d_in[2];  // (8,4096,16,32) f32
    const float* G     = (const float*)d_in[3];  // (8,4096,16) f32
    float* out = (float*)d_out;                  // (8,4096) f32

    const int total  = in_sizes[3] / DEPTH;      // B*T = 32768
    const int blocks = (total + BLOCK - 1) / BLOCK;
    dag_exec_kernel<<<blocks, BLOCK, 0, stream>>>(Vmag, Vsign, O, G, out, total);
}